// QuantizationModule_68650757259605
// MI455X (gfx1250) — compile-verified
//
#include <hip/hip_runtime.h>
#include <math.h>

// Problem constants (match reference)
constexpr int Bsz = 8, NF = 4096, INK = 512;
constexpr int NB = 2, NW = 320, CD = 384;
constexpr int NCOL = NB * NW;        // 640
constexpr int DM = NB * CD;          // 768
constexpr int NFRAMES = Bsz * NF;    // 32768

// Tiling
constexpr int TM   = 32;             // frames per block (2 row tiles of 16)
constexpr int NTHR = 512;            // 16 waves
constexpr int KC   = 128;            // K chunk staged in LDS
constexpr int LDA  = KC + 4;         // padded A stride (bank-conflict-free)
constexpr int LDL  = NCOL + 4;       // padded logits stride (644)

typedef __attribute__((ext_vector_type(2))) float v2f;
typedef __attribute__((ext_vector_type(8))) float v8f;

__global__ void vq_init(unsigned long long* marg) {
    for (int i = threadIdx.x; i < NCOL; i += 256) marg[i] = 0ull;
}

__global__ __launch_bounds__(NTHR) void vq_fused(
    const float* __restrict__ x,        // (B,NF,IN)
    const float* __restrict__ W,        // (IN, NCOL)
    const float* __restrict__ bias,     // (NCOL)
    const float* __restrict__ codebooks,// (NB,NW,CD)
    const float* __restrict__ gumbel,   // (B,NF,NB,NW)
    const int*   __restrict__ mask,     // (B,NF)  nonzero = invalid frame
    float* __restrict__ out,            // (B,NF,DM)
    unsigned long long* __restrict__ marg) // (NCOL) fixed-point 2^32 accum
{
    __shared__ float sA[TM * LDA];                  // 16896 B
    __shared__ float sL[TM * LDL];                  // 82432 B
    __shared__ unsigned long long sMarg[NCOL];      // 5120 B
    __shared__ int sIdx[TM * NB];                   // 256 B

    const int tid  = threadIdx.x;
    const int wave = tid >> 5;
    const int lane = tid & 31;
    const int ml   = lane & 15;      // M (for A) / N (for B) within tile
    const int mHalf = lane >> 4;     // 0: K+0/K+1 ; 1: K+2/K+3
    const int rt   = wave >> 3;      // row tile (0..1)
    const int cw   = wave & 7;       // column-wave (0..7), 5 col tiles each
    const int row0 = blockIdx.x * TM;

    for (int i = tid; i < NCOL; i += NTHR) sMarg[i] = 0ull;

    // ---- GEMM: each wave = 16x80 output tile, K in 4 chunks of 128 ----
    v8f acc[5];
#pragma unroll
    for (int t = 0; t < 5; ++t) acc[t] = (v8f){0.f,0.f,0.f,0.f,0.f,0.f,0.f,0.f};

    for (int kc = 0; kc < INK; kc += KC) {
        __syncthreads();
        // stage x tile chunk: 32 rows x 128 floats (float4, coalesced)
        for (int f = tid; f < TM * (KC / 4); f += NTHR) {
            int r  = f >> 5;          // 32 float4 per row
            int c4 = f & 31;
            float4 v = *(const float4*)(x + (size_t)(row0 + r) * INK + kc + c4 * 4);
            *(float4*)(&sA[r * LDA + c4 * 4]) = v;
        }
        __syncthreads();

#pragma unroll
        for (int t = 0; t < 5; ++t) {
            const int n0 = (cw * 5 + t) * 16;
            for (int kk = 0; kk < KC; kk += 4) {
                // A fragment (16x4 f32): lane half selects K+0/1 vs K+2/3
                v2f av;
                av.x = sA[(rt * 16 + ml) * LDA + kk + 2 * mHalf];
                av.y = sA[(rt * 16 + ml) * LDA + kk + 2 * mHalf + 1];
                // B fragment (4x16 f32) straight from global (L2-resident W)
                const float* wp = W + (size_t)(kc + kk + 2 * mHalf) * NCOL + n0 + ml;
                v2f bv;
                bv.x = wp[0];
                bv.y = wp[NCOL];
                acc[t] = __builtin_amdgcn_wmma_f32_16x16x4_f32(
                    false, av, false, bv, (short)0, acc[t], false, false);
            }
        }
    }
    __syncthreads();

    // ---- spill logits (+bias) to LDS: VGPR r -> row (mHalf?8+r:r), col n0+ml ----
#pragma unroll
    for (int t = 0; t < 5; ++t) {
        const int n0 = (cw * 5 + t) * 16;
        const float bv = bias[n0 + ml];
#pragma unroll
        for (int r = 0; r < 8; ++r) {
            int row = rt * 16 + (mHalf ? (8 + r) : r);
            sL[row * LDL + n0 + ml] = acc[t][r] + bv;
        }
    }
    __syncthreads();

    // ---- softmax / argmax: 64 problems (row,nb), one wave each, 4 per wave ----
    for (int p = wave; p < TM * NB; p += 16) {
        const int row = p >> 1, n = p & 1;
        const size_t gbase = (size_t)(row0 + row) * NCOL + (size_t)n * NW;
        float lv[10];
        float ymax = -INFINITY, lmax = -INFINITY;
        int yarg = 0;
#pragma unroll
        for (int j = 0; j < 10; ++j) {
            const int w = lane + 32 * j;
            const float l = sL[row * LDL + n * NW + w];
            const float g = gumbel[gbase + w];
            lv[j] = l;
            const float y = l + g;            // TAU == 1
            if (y > ymax) { ymax = y; yarg = w; }   // first-occurrence tie-break in-lane
            lmax = fmaxf(lmax, l);
        }
        // wave32 reduce: argmax (smaller index on tie) + max
#pragma unroll
        for (int off = 16; off > 0; off >>= 1) {
            float oy = __shfl_xor(ymax, off, 32);
            int   oa = __shfl_xor(yarg, off, 32);
            float om = __shfl_xor(lmax, off, 32);
            if (oy > ymax || (oy == ymax && oa < yarg)) { ymax = oy; yarg = oa; }
            lmax = fmaxf(lmax, om);
        }
        float ev[10];
        float s = 0.f;
#pragma unroll
        for (int j = 0; j < 10; ++j) { ev[j] = __expf(lv[j] - lmax); s += ev[j]; }
#pragma unroll
        for (int off = 16; off > 0; off >>= 1) s += __shfl_xor(s, off, 32);

        if (mask[row0 + row] == 0) {          // valid frame -> accumulate marginal
            const float inv = 1.f / s;
#pragma unroll
            for (int j = 0; j < 10; ++j) {
                unsigned long long fx =
                    (unsigned long long)(ev[j] * inv * 4294967296.0f); // 2^32 fixed point
                atomicAdd(&sMarg[n * NW + lane + 32 * j], fx);
            }
        }
        if (lane == 0) sIdx[row * NB + n] = yarg;
    }
    __syncthreads();

    // ---- flush per-block marginal (deterministic integer atomics) ----
    for (int i = tid; i < NCOL; i += NTHR) {
        unsigned long long v = sMarg[i];
        if (v) atomicAdd(&marg[i], v);
    }

    // ---- codebook gather: 64 (row,n) pairs x 96 float4 each ----
    for (int i = tid; i < TM * NB * (CD / 4); i += NTHR) {
        const int pr = i / (CD / 4), v4 = i % (CD / 4);
        const int row = pr >> 1, n = pr & 1;
        const int idx = sIdx[row * NB + n];
        float4 cv = *(const float4*)(codebooks + ((size_t)n * NW + idx) * CD + v4 * 4);
        *(float4*)(out + (size_t)(row0 + row) * DM + n * CD + v4 * 4) = cv;
    }
}

__global__ void vq_finalize(const unsigned long long* __restrict__ marg,
                            const int* __restrict__ mask,
                            float* __restrict__ outScalar)
{
    __shared__ int   scnt[256];
    __shared__ float sh0[256], sh1[256];
    const int tid = threadIdx.x;

    int c = 0;
    for (int i = tid; i < NFRAMES; i += 256) c += (mask[i] != 0);
    scnt[tid] = c;
    __syncthreads();
    for (int s = 128; s > 0; s >>= 1) {
        if (tid < s) scnt[tid] += scnt[tid + s];
        __syncthreads();
    }
    const float invN = 1.0f / (float)scnt[0];   // reference divides by mask.sum()

    const float scale = 1.0f / 4294967296.0f;
    float h0 = 0.f, h1 = 0.f;
    for (int i = tid; i < NCOL; i += 256) {
        float m = (float)marg[i] * scale * invN;
        float t = m * logf(m + 1e-7f);
        if (i < NW) h0 += t; else h1 += t;
    }
    sh0[tid] = h0; sh1[tid] = h1;
    __syncthreads();
    for (int s = 128; s > 0; s >>= 1) {
        if (tid < s) { sh0[tid] += sh0[tid + s]; sh1[tid] += sh1[tid + s]; }
        __syncthreads();
    }
    if (tid == 0) *outScalar = __expf(-sh0[0]) + __expf(-sh1[0]);
}

extern "C" void kernel_launch(void* const* d_in, const int* in_sizes, int n_in,
                              void* d_out, int out_size, void* d_ws, size_t ws_size,
                              hipStream_t stream) {
    const float* x    = (const float*)d_in[0];
    const float* W    = (const float*)d_in[1];
    const float* b    = (const float*)d_in[2];
    const float* cb   = (const float*)d_in[3];
    const float* gum  = (const float*)d_in[4];
    const int*   mask = (const int*)d_in[5];
    float* out = (float*)d_out;
    unsigned long long* marg = (unsigned long long*)d_ws;  // 640 * 8 B

    vq_init<<<1, 256, 0, stream>>>(marg);
    vq_fused<<<NFRAMES / TM, NTHR, 0, stream>>>(x, W, b, cb, gum, mask, out, marg);
    vq_finalize<<<1, 256, 0, stream>>>(marg, mask, out + (out_size - 1));
}